// ExtendedElectricalGridModel_11768210391596
// MI455X (gfx1250) — compile-verified
//
#include <hip/hip_runtime.h>
#include <cstdint>

// MI455X (gfx1250) GAT x2 + head.
// Dense transforms use V_WMMA_F32_16X16X4_F32 (fp32 WMMA, no precision loss).
// Edge phases use hardware no-return f32 atomics (L2-resident working set).

typedef __attribute__((ext_vector_type(2))) float v2f;
typedef __attribute__((ext_vector_type(8))) float v8f;

#define H_DIM 32
#define FIN   15

// float atomic-max via signed/unsigned integer atomics (monotone bit tricks).
__device__ __forceinline__ void atomic_max_f32(float* addr, float v) {
    if (v >= 0.0f) atomicMax((int*)addr, __float_as_int(v));
    else           atomicMin((unsigned int*)addr, __float_as_uint(v));
}

// ---------------------------------------------------------------- init
__global__ void k_init(float* deg, float* easum, float* m, float* z,
                       float* agg, int N) {
    int i = blockIdx.x * blockDim.x + threadIdx.x;
    if (i < N * H_DIM) agg[i] = 0.0f;
    if (i < N) { deg[i] = 0.0f; easum[i] = 0.0f; m[i] = -INFINITY; z[i] = 0.0f; }
}

// ------------------------------------------------- coefficient folding
// c_src = W^T a_src  (so s_src[n] = x[n] . c_src), coefE = a_e . We[:,0]
__global__ void k_precoef(const float* W1, const float* as1, const float* ad1,
                          const float* We1, const float* ae1,
                          const float* W2, const float* as2, const float* ad2,
                          const float* We2, const float* ae2, float* coef) {
    int t = threadIdx.x;
    if (t < 16) {                      // c_src1[0..14]
        float s = 0.0f;
        if (t < FIN) for (int h = 0; h < H_DIM; ++h) s += W1[h * FIN + t] * as1[h];
        coef[t] = s;
    } else if (t < 32) {               // c_dst1
        int k = t - 16; float s = 0.0f;
        if (k < FIN) for (int h = 0; h < H_DIM; ++h) s += W1[h * FIN + k] * ad1[h];
        coef[t] = s;
    } else if (t < 64) {               // c_src2[0..31]
        int k = t - 32; float s = 0.0f;
        for (int h = 0; h < H_DIM; ++h) s += W2[h * H_DIM + k] * as2[h];
        coef[t] = s;
    } else if (t < 96) {               // c_dst2
        int k = t - 64; float s = 0.0f;
        for (int h = 0; h < H_DIM; ++h) s += W2[h * H_DIM + k] * ad2[h];
        coef[t] = s;
    } else if (t == 96) {              // coefE1
        float s = 0.0f;
        for (int h = 0; h < H_DIM; ++h) s += We1[h] * ae1[h];
        coef[96] = s;
    } else if (t == 97) {              // coefE2
        float s = 0.0f;
        for (int h = 0; h < H_DIM; ++h) s += We2[h] * ae2[h];
        coef[97] = s;
    }
}

// -------------------------------------- degree + edge_attr sum per dst
__global__ void k_deg(const int* __restrict__ dst, const float* __restrict__ ea,
                      float* deg, float* easum, int E) {
    int e = blockIdx.x * blockDim.x + threadIdx.x;
    if (e >= E) return;
    int d = dst[e];
    unsafeAtomicAdd(&deg[d], 1.0f);
    unsafeAtomicAdd(&easum[d], ea[e]);
}

// --------------------------------------------- per-node logit scores
template <int KIN>
__global__ void k_score(const float* __restrict__ X,
                        const float* __restrict__ csrc,
                        const float* __restrict__ cdst,
                        float* ssrc, float* sdst, int N) {
    int n = blockIdx.x * blockDim.x + threadIdx.x;
    if (n >= N) return;
    const float* row = X + (size_t)n * KIN;
    float a = 0.0f, b = 0.0f;
#pragma unroll
    for (int k = 0; k < KIN; ++k) {
        float x = row[k];
        a += x * csrc[k];
        b += x * cdst[k];
    }
    ssrc[n] = a; sdst[n] = b;
}

// ---------------------------------------------------- fp32 WMMA GEMM
// out[N x HOUT] = (A[N x KIN] + ibias) @ W[HOUT x KIN]^T + obias
// One wave per 16-row tile (8 waves/block); K in chunks of 4 via
// V_WMMA_F32_16X16X4_F32. Fragment layouts per CDNA5 ISA:
//   A (16x4 f32): lane l holds row l&15, K = 2*(l>>4) + {0,1}
//   B (4x16 f32): lane holds col l&15, VGPR v holds K = 2*(l>>4) + v
//   C/D (16x16):  VGPR v, lane l -> D[v + 8*(l>>4)][l&15]
// All shapes are compile-time: K loop fully unrolls, loads are unconditional
// (indices clamped in-bounds; out-of-range values zeroed via lane selects),
// stores take a wave-uniform full-tile fast path.
template <int KIN, int HOUT, int NCOLS, int LDO, bool HAS_IB, bool HAS_OB>
__global__ void k_wmma_xform(const float* __restrict__ A,
                             const float* __restrict__ W,
                             const float* __restrict__ ibias,
                             const float* __restrict__ obias,
                             float* __restrict__ out, int Nrows) {
    constexpr int KC = (KIN + 3) / 4;            // K chunks of 4
    constexpr int NT = (HOUT + 15) / 16;         // 16-col output tiles
    constexpr bool COLGUARD = (NCOLS < NT * 16); // ragged last column tile?
    const int lane = threadIdx.x & 31;
    const int wave = threadIdx.x >> 5;
    const int half = lane >> 4;
    const int l16  = lane & 15;
    const int tile = blockIdx.x * (blockDim.x >> 5) + wave;
    const int row0 = tile * 16;
    if (row0 >= Nrows) return;

    // Unconditional A loads: clamp row (clamped rows never stored).
    const int r = row0 + l16;
    const int rload = (r < Nrows) ? r : (Nrows - 1);
    const float* Arow = A + (size_t)rload * KIN;

    // A fragments for every K chunk, reused across all column tiles.
    v2f afrag[KC];
#pragma unroll
    for (int kc = 0; kc < KC; ++kc) {
        const int kb = kc * 4 + half * 2;
        const int k0 = (kb < KIN) ? kb : (KIN - 1);         // clamp (in-bounds)
        const int k1 = (kb + 1 < KIN) ? kb + 1 : (KIN - 1);
        float ax = Arow[k0];
        float ay = Arow[k1];
        if (HAS_IB) { ax += ibias[k0]; ay += ibias[k1]; }
        if (kb >= KIN)     ax = 0.0f;                       // lane select, no branch
        if (kb + 1 >= KIN) ay = 0.0f;
        afrag[kc].x = ax;
        afrag[kc].y = ay;
    }

    const bool fullTile = (row0 + 16 <= Nrows);             // wave-uniform

#pragma unroll
    for (int jt = 0; jt < NT; ++jt) {
        const int colB  = jt * 16 + l16;
        const int cload = (colB < HOUT) ? colB : (HOUT - 1);
        v8f acc = {0.f, 0.f, 0.f, 0.f, 0.f, 0.f, 0.f, 0.f};
#pragma unroll
        for (int kc = 0; kc < KC; ++kc) {
            const int kb = kc * 4 + half * 2;
            const int k0 = (kb < KIN) ? kb : (KIN - 1);
            const int k1 = (kb + 1 < KIN) ? kb + 1 : (KIN - 1);
            float bx = W[(size_t)cload * KIN + k0];
            float by = W[(size_t)cload * KIN + k1];
            if (kb >= KIN || colB >= HOUT)     bx = 0.0f;   // pad K / pad cols
            if (kb + 1 >= KIN || colB >= HOUT) by = 0.0f;
            v2f b; b.x = bx; b.y = by;
            acc = __builtin_amdgcn_wmma_f32_16x16x4_f32(
                /*neg_a=*/false, afrag[kc], /*neg_b=*/false, b,
                /*c_mod=*/(short)0, acc, /*reuse_a=*/false, /*reuse_b=*/false);
        }
        const int col = jt * 16 + l16;
        float ob = 0.0f;
        if (HAS_OB) ob = obias[(col < NCOLS) ? col : (NCOLS - 1)];
        const int base = (row0 + half * 8) * LDO + col;     // 32-bit addressing
        if (fullTile) {
#pragma unroll
            for (int v = 0; v < 8; ++v) {
                if (!COLGUARD || col < NCOLS) {
                    float val = acc[v];
                    if (HAS_OB) val += ob;                  // no +0.0 junk otherwise
                    out[base + v * LDO] = val;
                }
            }
        } else {                                            // ragged row tail
#pragma unroll
            for (int v = 0; v < 8; ++v) {
                const int rr = row0 + v + half * 8;
                if (rr < Nrows && (!COLGUARD || col < NCOLS)) {
                    float val = acc[v];
                    if (HAS_OB) val += ob;
                    out[rr * LDO + col] = val;
                }
            }
        }
    }
}

// ------------------------------------- edge pass 1: logits + seg-max
__global__ void k_edge_max(const int* __restrict__ src, const int* __restrict__ dst,
                           const float* __restrict__ ea,
                           const float* __restrict__ deg, const float* __restrict__ easum,
                           const float* __restrict__ ssrc, const float* __restrict__ sdst,
                           const float* __restrict__ coefE,
                           float* tmp, float* m, int E, int N) {
    int e = blockIdx.x * blockDim.x + threadIdx.x;
    if (e >= E + N) return;
    int s, d; float a;
    if (e < E) { s = src[e]; d = dst[e]; a = ea[e]; }
    else       { s = d = e - E; a = easum[d] / fmaxf(deg[d], 1.0f); }  // mean fill
    float v = ssrc[s] + sdst[d] + coefE[0] * a;
    v = (v < 0.0f) ? 0.2f * v : v;            // leaky_relu(., 0.2)
    tmp[e] = v;
    atomic_max_f32(&m[d], v);
}

// ------------------------------------- edge pass 2: exp + seg-sum
__global__ void k_edge_expsum(const int* __restrict__ dst, float* tmp,
                              const float* __restrict__ m, float* z, int E, int N) {
    int e = blockIdx.x * blockDim.x + threadIdx.x;
    if (e >= E + N) return;
    int d = (e < E) ? dst[e] : (e - E);
    float v = __expf(tmp[e] - m[d]);
    tmp[e] = v;
    unsafeAtomicAdd(&z[d], v);
}

// ----------------------- edge pass 3: normalize + weighted scatter-add
// 4 lanes per edge, 8 features each (2x float4 gather, 8 f32 atomics).
__global__ void k_edge_agg(const int* __restrict__ src, const int* __restrict__ dst,
                           const float* __restrict__ tmp, const float* __restrict__ z,
                           const float* __restrict__ h, float* agg, int E, int N) {
    long long t = (long long)blockIdx.x * blockDim.x + threadIdx.x;
    long long e = t >> 2;
    if (e >= (long long)E + N) return;
    int part = (int)(t & 3);
    int s, d;
    if (e < (long long)E) { s = src[e]; d = dst[e]; }
    else                  { s = d = (int)(e - E); }
    float w = tmp[e] / z[d];
    const float4* hs = (const float4*)(h + (size_t)s * H_DIM + part * 8);
    float4 h0 = hs[0], h1 = hs[1];
    float* ad = agg + (size_t)d * H_DIM + part * 8;
    unsafeAtomicAdd(ad + 0, w * h0.x);
    unsafeAtomicAdd(ad + 1, w * h0.y);
    unsafeAtomicAdd(ad + 2, w * h0.z);
    unsafeAtomicAdd(ad + 3, w * h0.w);
    unsafeAtomicAdd(ad + 4, w * h1.x);
    unsafeAtomicAdd(ad + 5, w * h1.y);
    unsafeAtomicAdd(ad + 6, w * h1.z);
    unsafeAtomicAdd(ad + 7, w * h1.w);
}

// ------------- bias+ReLU into hrelu; recycle agg/m/z for layer 2
__global__ void k_bias_relu_reset(float* agg, const float* __restrict__ b1,
                                  float* hrelu, float* m, float* z, int N) {
    int i = blockIdx.x * blockDim.x + threadIdx.x;
    if (i < N * H_DIM) {
        float v = agg[i] + b1[i & (H_DIM - 1)];
        agg[i] = 0.0f;
        hrelu[i] = fmaxf(v, 0.0f);
    }
    if (i < N) { m[i] = -INFINITY; z[i] = 0.0f; }
}

// ================================================================
extern "C" void kernel_launch(void* const* d_in, const int* in_sizes, int n_in,
                              void* d_out, int out_size, void* d_ws, size_t ws_size,
                              hipStream_t stream) {
    const float* x   = (const float*)d_in[0];
    const int*   ei  = (const int*)  d_in[1];
    const float* ea  = (const float*)d_in[2];
    const float* W1  = (const float*)d_in[3];
    const float* as1 = (const float*)d_in[4];
    const float* ad1 = (const float*)d_in[5];
    const float* We1 = (const float*)d_in[6];
    const float* ae1 = (const float*)d_in[7];
    const float* b1  = (const float*)d_in[8];
    const float* W2  = (const float*)d_in[9];
    const float* as2 = (const float*)d_in[10];
    const float* ad2 = (const float*)d_in[11];
    const float* We2 = (const float*)d_in[12];
    const float* ae2 = (const float*)d_in[13];
    const float* b2  = (const float*)d_in[14];
    const float* Wl  = (const float*)d_in[15];
    const float* bl  = (const float*)d_in[16];

    const int N = in_sizes[0] / FIN;     // 100000
    const int E = in_sizes[2];           // 3200000 (edge_attr, ED=1)
    const int EN = E + N;                // with self-loops
    const int* src = ei;
    const int* dst = ei + E;

    // workspace layout (floats)
    float* ws    = (float*)d_ws;
    float* deg   = ws;
    float* easum = ws + (size_t)N;
    float* ssrc  = ws + (size_t)2 * N;
    float* sdst  = ws + (size_t)3 * N;
    float* m     = ws + (size_t)4 * N;
    float* z     = ws + (size_t)5 * N;
    float* coef  = ws + (size_t)6 * N;              // 128 slots
    float* h     = coef + 128;                      // N*32
    float* hrelu = h + (size_t)N * H_DIM;           // N*32
    float* agg   = hrelu + (size_t)N * H_DIM;       // N*32
    float* tmp   = agg + (size_t)N * H_DIM;         // E+N

    const int B = 256;
    const int gNH  = (N * H_DIM + B - 1) / B;
    const int gN   = (N + B - 1) / B;
    const int gE   = (E + B - 1) / B;
    const int gEN  = (EN + B - 1) / B;
    const int gEN4 = (int)(((long long)EN * 4 + B - 1) / B);
    const int nTiles = (N + 15) / 16;               // 16-row WMMA tiles
    const int gT   = (nTiles + 7) / 8;              // 8 waves (tiles) per block

    k_init<<<gNH, B, 0, stream>>>(deg, easum, m, z, agg, N);
    k_precoef<<<1, 128, 0, stream>>>(W1, as1, ad1, We1, ae1,
                                     W2, as2, ad2, We2, ae2, coef);
    k_deg<<<gE, B, 0, stream>>>(dst, ea, deg, easum, E);

    // ---------------- layer 1 ----------------
    k_score<FIN><<<gN, B, 0, stream>>>(x, coef, coef + 16, ssrc, sdst, N);
    k_wmma_xform<FIN, H_DIM, H_DIM, H_DIM, false, false>
        <<<gT, B, 0, stream>>>(x, W1, nullptr, nullptr, h, N);
    k_edge_max<<<gEN, B, 0, stream>>>(src, dst, ea, deg, easum, ssrc, sdst,
                                      coef + 96, tmp, m, E, N);
    k_edge_expsum<<<gEN, B, 0, stream>>>(dst, tmp, m, z, E, N);
    k_edge_agg<<<gEN4, B, 0, stream>>>(src, dst, tmp, z, h, agg, E, N);
    k_bias_relu_reset<<<gNH, B, 0, stream>>>(agg, b1, hrelu, m, z, N);

    // ---------------- layer 2 ----------------
    k_score<H_DIM><<<gN, B, 0, stream>>>(hrelu, coef + 32, coef + 64, ssrc, sdst, N);
    k_wmma_xform<H_DIM, H_DIM, H_DIM, H_DIM, false, false>
        <<<gT, B, 0, stream>>>(hrelu, W2, nullptr, nullptr, h, N);
    k_edge_max<<<gEN, B, 0, stream>>>(src, dst, ea, deg, easum, ssrc, sdst,
                                      coef + 97, tmp, m, E, N);
    k_edge_expsum<<<gEN, B, 0, stream>>>(dst, tmp, m, z, E, N);
    k_edge_agg<<<gEN4, B, 0, stream>>>(src, dst, tmp, z, h, agg, E, N);

    // ---------------- head: out = (agg + b2) @ Wl^T + bl ----------------
    k_wmma_xform<H_DIM, 20, 20, 20, true, true>
        <<<gT, B, 0, stream>>>(agg, Wl, b2, bl, (float*)d_out, N);
}